// CAGPool_layer_50981261804244
// MI455X (gfx1250) — compile-verified
//
#include <hip/hip_runtime.h>
#include <hip/hip_bf16.h>

typedef __attribute__((ext_vector_type(2))) float v2f;
typedef __attribute__((ext_vector_type(4))) float v4f;
typedef __attribute__((ext_vector_type(8))) float v8f;

#define NB    64                 // graphs
#define NPG   2000               // nodes per graph
#define NC    128                // channels
#define NE    2048000            // edges
#define NTOT  (NB * NPG)         // 128000 nodes
#define KTOP  1000               // ceil(0.5 * NPG)
#define BK    (NB * KTOP)        // 64000 selected nodes
#define CPG   (NPG / 16)         // 125 chunks per graph
#define NCHUNK (NTOT / 16)       // 8000 16-node chunks
#define SORT_N 2048

// ---------------------------------------------------------------------------
// 1) inv_norm[g] = 1 / ||pool_vector[g]||   (one wave per graph)
// ---------------------------------------------------------------------------
__global__ void cag_norm_kernel(const float* __restrict__ pv,
                                float* __restrict__ inv_norm) {
    int g = blockIdx.x;
    int lane = threadIdx.x;                       // 32 threads
    v4f v = *(const v4f*)(pv + g * NC + lane * 4);
    float s = v.x * v.x + v.y * v.y + v.z * v.z + v.w * v.w;
    #pragma unroll
    for (int off = 16; off > 0; off >>= 1) s += __shfl_xor(s, off, 32);
    if (lane == 0) inv_norm[g] = rsqrtf(s);
}

// ---------------------------------------------------------------------------
// 2) score[n] = dot(x[n], pv[g]) * inv_norm[g] via V_WMMA_F32_16X16X4_F32.
//    One wave per 16-node chunk; A = 16x4 tile of x, B = pv K-slice in all
//    16 columns, 32 accumulating WMMA steps over K=128. D replicates each
//    node's dot product across N; lanes 0 / 16 extract 8 scores each.
// ---------------------------------------------------------------------------
__global__ void cag_score_kernel(const float* __restrict__ x,
                                 const float* __restrict__ pv,
                                 const float* __restrict__ inv_norm,
                                 float* __restrict__ score) {
    int wid  = blockIdx.x * (blockDim.x >> 5) + (threadIdx.x >> 5); // chunk id
    int lane = threadIdx.x & 31;
    if (wid >= NCHUNK) return;                    // wave-uniform (never taken)
    int g   = wid / CPG;
    int row = lane & 15;                          // A-matrix M = row
    int kh  = (lane >> 4) << 1;                   // lanes 0-15: K+{0,1}; 16-31: K+{2,3}
    const float* xr = x + (size_t)(wid * 16 + row) * NC + kh;
    const float* pr = pv + g * NC + kh;

    v8f acc = {};
    #pragma unroll 8
    for (int k0 = 0; k0 < NC; k0 += 4) {
        v2f a = *(const v2f*)(xr + k0);           // {A[M][k], A[M][k+1]}
        v2f b = *(const v2f*)(pr + k0);           // pv slice, all columns equal
        acc = __builtin_amdgcn_wmma_f32_16x16x4_f32(
            /*neg_a=*/false, a, /*neg_b=*/false, b,
            /*c_mod=*/(short)0, acc, /*reuse_a=*/false, /*reuse_b=*/false);
    }

    float inv = inv_norm[g];
    if (lane == 0) {                              // D: M = 0..7  at N=0
        #pragma unroll
        for (int v = 0; v < 8; ++v) score[wid * 16 + v] = acc[v] * inv;
    } else if (lane == 16) {                      // D: M = 8..15 at N=0
        #pragma unroll
        for (int v = 0; v < 8; ++v) score[wid * 16 + 8 + v] = acc[v] * inv;
    }
}

// ---------------------------------------------------------------------------
// 3) Per-graph stable descending argsort (bitonic, 2048 u64 keys in LDS),
//    emit perm / new_batch / mask for the top KTOP nodes.
// ---------------------------------------------------------------------------
__global__ __launch_bounds__(1024) void cag_sort_kernel(
        const float* __restrict__ score, int* __restrict__ mask,
        int* __restrict__ perm_out, int* __restrict__ nbatch_out) {
    __shared__ unsigned long long keys[SORT_N];   // 16 KB
    int g = blockIdx.x;
    int tid = threadIdx.x;

    for (int i = tid; i < SORT_N; i += 1024) {
        unsigned long long key;
        if (i < NPG) {
            unsigned u = __float_as_uint(score[g * NPG + i]);
            u = (u & 0x80000000u) ? ~u : (u | 0x80000000u); // order-preserving
            key = ((unsigned long long)(~u) << 32) | (unsigned)i; // desc score, asc idx
        } else {
            key = ~0ull;                          // padding sorts last
        }
        keys[i] = key;
    }
    __syncthreads();

    for (int size = 2; size <= SORT_N; size <<= 1) {
        for (int stride = size >> 1; stride > 0; stride >>= 1) {
            int lo = 2 * tid - (tid & (stride - 1));
            int hi = lo + stride;
            bool desc = (lo & size) != 0;
            unsigned long long a = keys[lo], b = keys[hi];
            if ((a > b) != desc) { keys[lo] = b; keys[hi] = a; }
            __syncthreads();
        }
    }

    // mask init for this graph's node range, then scatter selected indices
    for (int i = tid; i < NPG; i += 1024) mask[g * NPG + i] = -1;
    __threadfence_block();
    __syncthreads();

    if (tid < KTOP) {
        int li   = (int)(keys[tid] & 0xFFFFFFFFull);
        int node = g * NPG + li;
        int ridx = g * KTOP + tid;
        perm_out[ridx]   = node;
        nbatch_out[ridx] = g;
        mask[node]       = ridx;
    }
}

// ---------------------------------------------------------------------------
// 4) x_out[r] = x[perm[r]] * sigmoid(score[perm[r]])  (one wave per row)
// ---------------------------------------------------------------------------
__global__ void cag_gather_kernel(const float* __restrict__ x,
                                  const float* __restrict__ score,
                                  const int* __restrict__ perm,
                                  float* __restrict__ xout) {
    int wid  = blockIdx.x * (blockDim.x >> 5) + (threadIdx.x >> 5);
    int lane = threadIdx.x & 31;
    if (wid >= BK) return;
    int p = perm[wid];
    float s   = score[p];
    float sig = 1.0f / (1.0f + __expf(-s));
    v4f xv = *(const v4f*)(x + (size_t)p * NC + lane * 4);
    xv *= sig;
    *(v4f*)(xout + (size_t)wid * NC + lane * 4) = xv;
}

// ---------------------------------------------------------------------------
// 5) Edge relabel/filter (one thread per edge; mask is L2-resident)
// ---------------------------------------------------------------------------
__global__ void cag_edge_kernel(const int* __restrict__ ei,
                                const int* __restrict__ mask,
                                int* __restrict__ ei_out,
                                int* __restrict__ valid_out) {
    int e = blockIdx.x * blockDim.x + threadIdx.x;
    if (e >= NE) return;
    int r = mask[ei[e]];
    int c = mask[ei[NE + e]];
    bool ok = (r >= 0) && (c >= 0);
    ei_out[e]      = ok ? r : -1;
    ei_out[NE + e] = ok ? c : -1;
    valid_out[e]   = ok ? 1 : 0;
}

// ---------------------------------------------------------------------------
extern "C" void kernel_launch(void* const* d_in, const int* in_sizes, int n_in,
                              void* d_out, int out_size, void* d_ws, size_t ws_size,
                              hipStream_t stream) {
    const float* x   = (const float*)d_in[0];   // [NTOT, NC]
    const int*   ei  = (const int*)  d_in[1];   // [2, NE]
    const int*   bat = (const int*)  d_in[2];   // [NTOT] (layout implies it)
    const float* pv  = (const float*)d_in[3];   // [NB, NC]
    const int*   csz = (const int*)  d_in[4];   // [NB]  (constant NPG)
    (void)bat; (void)csz; (void)in_sizes; (void)n_in; (void)out_size; (void)ws_size;

    // workspace: score[NTOT] f32 | inv_norm[NB] f32 | mask[NTOT] i32
    float* ws       = (float*)d_ws;
    float* score    = ws;
    float* inv_norm = ws + NTOT;
    int*   mask     = (int*)(ws + NTOT + NB);

    // outputs flat in return order (4-byte elements)
    float* xout       = (float*)d_out;                         // BK*NC
    int*   ei_out     = (int*)((float*)d_out + (size_t)BK*NC); // 2*NE
    int*   valid_out  = ei_out + 2 * (size_t)NE;               // NE
    int*   nbatch_out = valid_out + NE;                        // BK
    int*   perm_out   = nbatch_out + BK;                       // BK

    cag_norm_kernel <<<NB, 32, 0, stream>>>(pv, inv_norm);
    cag_score_kernel<<<NCHUNK / 8, 256, 0, stream>>>(x, pv, inv_norm, score);
    cag_sort_kernel <<<NB, 1024, 0, stream>>>(score, mask, perm_out, nbatch_out);
    cag_gather_kernel<<<BK / 8, 256, 0, stream>>>(x, score, perm_out, xout);
    cag_edge_kernel <<<(NE + 255) / 256, 256, 0, stream>>>(ei, mask, ei_out, valid_out);
}